// PointPillarScatter3d_57724360458630
// MI455X (gfx1250) — compile-verified
//
#include <hip/hip_runtime.h>
#include <stdint.h>

// PointPillarScatter3d for MI455X (gfx1250).
// Strategy: invert the scatter into a gather.
//   1) init a cell->pillar index map (B*NZ*NY*NX int32 = 4 MiB, lives in d_ws / L2)
//   2) scatter pillar ids into the map (320k cheap 4B stores)
//   3) streaming gather kernel: per block, stage one 1024-cell map chunk into LDS
//      (async-to-LDS), loop all 128 channels, gather features (L2/WGP$ resident),
//      write output fully coalesced with non-temporal b128 stores.

typedef float v4f __attribute__((ext_vector_type(4)));
typedef int   v4i __attribute__((ext_vector_type(4)));
typedef __attribute__((address_space(1))) v4i g_v4i;   // global (AS1) vector ptr
typedef __attribute__((address_space(3))) v4i s_v4i;   // LDS (AS3) vector ptr

#define NXC 512
#define NYC 512
#define NZC 1
#define SPB (NXC * NYC * NZC)   // cells per batch = 262144
#define CB  128                 // channels
#define CHUNK 1024              // cells per block in gather
#define GBLK 256                // threads per gather block (8 waves, wave32)

// ---------------- map init: map[i] = -1 ----------------
__global__ __launch_bounds__(256) void pp_init_map(int* __restrict__ map, int n4) {
    int i = blockIdx.x * blockDim.x + threadIdx.x;
    if (i < n4) {
        v4i m = {-1, -1, -1, -1};
        ((v4i*)map)[i] = m;
    }
}

// ---------------- scatter pillar ids into map ----------------
__global__ __launch_bounds__(256) void pp_scatter_ids(const int* __restrict__ coords,
                                                      int* __restrict__ map, int P) {
    int p = blockIdx.x * blockDim.x + threadIdx.x;
    if (p >= P) return;
    v4i c = ((const v4i*)coords)[p];             // (b, z, y, x)
    int cell = c.y * (NYC * NXC) + c.z * NXC + c.w;
    map[(size_t)c.x * SPB + cell] = p;
}

// ---------------- gather: out[(b*CB+c)*SPB + cell] ----------------
__global__ __launch_bounds__(GBLK) void pp_gather(const float* __restrict__ feats,
                                                  const int* __restrict__ map,
                                                  float* __restrict__ out) {
    __shared__ alignas(16) int smap[CHUNK];
    const int chunks_per_batch = SPB / CHUNK;                 // 256
    const int b    = blockIdx.x / chunks_per_batch;
    const int base = (blockIdx.x % chunks_per_batch) * CHUNK;
    const int tid  = threadIdx.x;
    const int* gmap = map + (size_t)b * SPB + base;

    // Stage the 4KB map chunk into LDS. Prefer the CDNA5 async-to-LDS path.
#if __has_builtin(__builtin_amdgcn_global_load_async_to_lds_b128)
    {
        // global pointers share bit representation between generic and AS1
        g_v4i* gsrc = (g_v4i*)(uintptr_t)(gmap + 4 * tid);
        s_v4i* ldst = (s_v4i*)(smap + 4 * tid);
        __builtin_amdgcn_global_load_async_to_lds_b128(gsrc, ldst, 0, 0);
    }
#if __has_builtin(__builtin_amdgcn_s_wait_asynccnt)
    __builtin_amdgcn_s_wait_asynccnt(0);
#else
    asm volatile("s_wait_asynccnt 0" ::: "memory");
#endif
#else
    ((v4i*)smap)[tid] = ((const v4i*)gmap)[tid];
#endif
    __syncthreads();

    const v4i pid = ((const v4i*)smap)[tid];     // 4 cells per thread

    // Clamp base pointers so loads are always in-bounds; select-with-0 afterwards.
    const float* f0 = feats + (size_t)(pid.x < 0 ? 0 : pid.x) * CB;
    const float* f1 = feats + (size_t)(pid.y < 0 ? 0 : pid.y) * CB;
    const float* f2 = feats + (size_t)(pid.z < 0 ? 0 : pid.z) * CB;
    const float* f3 = feats + (size_t)(pid.w < 0 ? 0 : pid.w) * CB;

    // Warm the gathered feature rows (global_prefetch_b8).
    if (pid.x >= 0) __builtin_prefetch(f0, 0, 1);
    if (pid.y >= 0) __builtin_prefetch(f1, 0, 1);
    if (pid.z >= 0) __builtin_prefetch(f2, 0, 1);
    if (pid.w >= 0) __builtin_prefetch(f3, 0, 1);

    float* outp = out + (size_t)b * CB * SPB + base + 4 * tid;

#pragma unroll 4
    for (int c = 0; c < CB; ++c) {
        v4f v;
        v.x = (pid.x >= 0) ? f0[c] : 0.0f;
        v.y = (pid.y >= 0) ? f1[c] : 0.0f;
        v.z = (pid.z >= 0) ? f2[c] : 0.0f;
        v.w = (pid.w >= 0) ? f3[c] : 0.0f;
        // 537MB streaming output: non-temporal so the feature table stays in L2
        __builtin_nontemporal_store(v, (v4f*)(outp + (size_t)c * SPB));
    }
}

// ---------------- fallback path (workspace too small) ----------------
__global__ __launch_bounds__(256) void pp_zero(float* __restrict__ out, long long n4) {
    long long i = blockIdx.x * (long long)blockDim.x + threadIdx.x;
    if (i < n4) {
        v4f z = {0.0f, 0.0f, 0.0f, 0.0f};
        __builtin_nontemporal_store(z, ((v4f*)out) + i);
    }
}

__global__ __launch_bounds__(256) void pp_direct(const float* __restrict__ feats,
                                                 const int* __restrict__ coords,
                                                 float* __restrict__ out, int P) {
    long long t = blockIdx.x * (long long)blockDim.x + threadIdx.x;
    if (t >= (long long)P * CB) return;
    int p = (int)(t >> 7);
    int c = (int)(t & (CB - 1));
    v4i co = ((const v4i*)coords)[p];
    size_t cell = (size_t)co.y * (NYC * NXC) + (size_t)co.z * NXC + (size_t)co.w;
    out[((size_t)co.x * CB + c) * SPB + cell] = feats[(size_t)p * CB + c];
}

extern "C" void kernel_launch(void* const* d_in, const int* in_sizes, int n_in,
                              void* d_out, int out_size, void* d_ws, size_t ws_size,
                              hipStream_t stream) {
    // inputs: [0]=batch_size (scalar, unused; derived from out_size),
    //         [1]=pillar_features (P*128 f32), [2]=coords (P*4 i32)
    const float* feats  = (const float*)d_in[1];
    const int*   coords = (const int*)d_in[2];
    float*       out    = (float*)d_out;

    const int P   = in_sizes[2] / 4;
    const int Bsz = out_size / (CB * SPB);
    const size_t map_bytes = (size_t)Bsz * SPB * sizeof(int);

    if (Bsz > 0 && ws_size >= map_bytes) {
        int* map = (int*)d_ws;
        const int n4 = (Bsz * SPB) / 4;
        pp_init_map<<<(n4 + 255) / 256, 256, 0, stream>>>(map, n4);
        pp_scatter_ids<<<(P + 255) / 256, 256, 0, stream>>>(coords, map, P);
        const int nblk = Bsz * (SPB / CHUNK);
        pp_gather<<<nblk, GBLK, 0, stream>>>(feats, map, out);
    } else {
        const long long n4 = (long long)out_size / 4;
        pp_zero<<<(int)((n4 + 255) / 256), 256, 0, stream>>>(out, n4);
        const long long nt = (long long)P * CB;
        pp_direct<<<(int)((nt + 255) / 256), 256, 0, stream>>>(feats, coords, out, P);
    }
}